// EventDrivenProcessor_21509196218754
// MI455X (gfx1250) — compile-verified
//
#include <hip/hip_runtime.h>
#include <hip/hip_bf16.h>

typedef __attribute__((ext_vector_type(2))) float v2f;
typedef __attribute__((ext_vector_type(8))) float v8f;

#define BDIM   384
#define TDIM   1024
#define DDIM   384
#define WWIN   100
#define THRESHV 0.05f

// One workgroup (256 threads = 8 wave32) per timestep t.
// Phase A: load pattern row pmem[t%W] into LDS.
// Phase B: scores[b] = spike[b,t,:] . row  via V_WMMA_F32_16X16X4_F32
//          (B operand = row chunk broadcast to all 16 columns; every output
//           column equals the score vector; extract column N=0).
//          Also accumulates has_event = any(spike[:,t,:] > 0).
// Phase C: softmax over the 384 scores in LDS.
// Phase D: out = thresh(tanh(0.7*spike + 0.3*pat_d)), float4-vectorized;
//          spike slice re-read hits L2 (589 KB/t slice, 192 MB L2).
__launch_bounds__(256)
__global__ void event_proc_kernel(const float* __restrict__ spike,
                                  const float* __restrict__ pmem,
                                  float* __restrict__ out) {
  const int t    = blockIdx.x;
  const int tid  = threadIdx.x;
  const int lane = tid & 31;
  const int wave = tid >> 5;

  __shared__ float r_lds[DDIM];   // pattern row for this t
  __shared__ float s_lds[BDIM];   // scores -> softmax probabilities
  __shared__ float red[256];      // reduction scratch
  __shared__ int   ev_flag;

  // ---- Phase A: pattern row -> LDS ----
  {
    const float* rp = pmem + (size_t)(t % WWIN) * DDIM;
    for (int i = tid; i < DDIM; i += 256) r_lds[i] = rp[i];
    if (tid == 0) ev_flag = 0;
  }
  __syncthreads();

  // ---- Phase B: WMMA scoring + has_event scan ----
  // A-frag layout (32-bit A 16x4): lanes 0-15 hold M=0..15 / K=0,1 ;
  // lanes 16-31 hold M=0..15 / K=2,3. B-frag symmetric over N.
  bool any_pos = false;
  const int half = lane >> 4;        // 0: K = d0,d0+1   1: K = d0+2,d0+3
  const int m    = lane & 15;        // row within tile
  #pragma unroll
  for (int tile = 0; tile < 3; ++tile) {
    const int b0 = (wave * 3 + tile) * 16;   // 8 waves * 3 tiles = 24 tiles = 384 rows
    v8f acc0 = {};
    v8f acc1 = {};
    const float* base = spike + ((size_t)(b0 + m) * TDIM + t) * DDIM + 2 * half;
    for (int d0 = 0; d0 < DDIM; d0 += 8) {
      const float2 av0 = *(const float2*)(base + d0);
      const float2 av1 = *(const float2*)(base + d0 + 4);
      any_pos = any_pos | (av0.x > 0.f) | (av0.y > 0.f) | (av1.x > 0.f) | (av1.y > 0.f);
      v2f a0; a0.x = av0.x; a0.y = av0.y;
      v2f a1; a1.x = av1.x; a1.y = av1.y;
      const int rb = d0 + 2 * half;
      v2f bb0; bb0.x = r_lds[rb];     bb0.y = r_lds[rb + 1];
      v2f bb1; bb1.x = r_lds[rb + 4]; bb1.y = r_lds[rb + 5];
      acc0 = __builtin_amdgcn_wmma_f32_16x16x4_f32(false, a0, false, bb0,
                                                   (short)0, acc0, false, false);
      acc1 = __builtin_amdgcn_wmma_f32_16x16x4_f32(false, a1, false, bb1,
                                                   (short)0, acc1, false, false);
    }
    v8f acc = acc0 + acc1;
    // C/D layout: VGPR j, lanes 0-15 -> (M=j,   N=lane),
    //                     lanes 16-31 -> (M=j+8, N=lane-16). Take N==0.
    if (lane == 0) {
      #pragma unroll
      for (int j = 0; j < 8; ++j) s_lds[b0 + j] = acc[j];
    }
    if (lane == 16) {
      #pragma unroll
      for (int j = 0; j < 8; ++j) s_lds[b0 + 8 + j] = acc[j];
    }
  }
  if (any_pos) ev_flag = 1;   // benign same-value race
  __syncthreads();

  // ---- Phase C: softmax over batch axis (384 values) ----
  float mv = -3.4e38f;
  for (int i = tid; i < BDIM; i += 256) mv = fmaxf(mv, s_lds[i]);
  red[tid] = mv;
  __syncthreads();
  for (int off = 128; off > 0; off >>= 1) {
    if (tid < off) red[tid] = fmaxf(red[tid], red[tid + off]);
    __syncthreads();
  }
  const float maxv = red[0];
  __syncthreads();

  float ps = 0.f;
  for (int i = tid; i < BDIM; i += 256) {   // disjoint per-thread subsets
    const float e = __expf(s_lds[i] - maxv);
    s_lds[i] = e;
    ps += e;
  }
  red[tid] = ps;
  __syncthreads();
  for (int off = 128; off > 0; off >>= 1) {
    if (tid < off) red[tid] += red[tid + off];
    __syncthreads();
  }
  const float inv = 1.0f / red[0];
  const bool  ev  = (ev_flag != 0);
  __syncthreads();
  for (int i = tid; i < BDIM; i += 256) s_lds[i] *= inv;
  __syncthreads();

  // ---- Phase D: fused elementwise output (float4 vectorized) ----
  // pat_term[b,t,d] = pattern[d,t] (B==D broadcast quirk), zeroed for t<W.
  const float  pscale = (t >= WWIN) ? 0.3f : 0.0f;
  const size_t tOff   = (size_t)t * DDIM;
  for (int b = wave; b < BDIM; b += 8) {
    const size_t rowOff = (size_t)b * TDIM * DDIM + tOff;
    for (int d4 = lane; d4 < DDIM / 4; d4 += 32) {
      const int    d   = d4 * 4;
      const size_t off = rowOff + d;
      const float4 sv  = *(const float4*)(spike + off);
      float4 ov;
      ov.x = tanhf(0.7f * sv.x + pscale * s_lds[d + 0]);
      ov.y = tanhf(0.7f * sv.y + pscale * s_lds[d + 1]);
      ov.z = tanhf(0.7f * sv.z + pscale * s_lds[d + 2]);
      ov.w = tanhf(0.7f * sv.w + pscale * s_lds[d + 3]);
      ov.x = (ev && fabsf(ov.x) > THRESHV) ? ov.x : 0.f;
      ov.y = (ev && fabsf(ov.y) > THRESHV) ? ov.y : 0.f;
      ov.z = (ev && fabsf(ov.z) > THRESHV) ? ov.z : 0.f;
      ov.w = (ev && fabsf(ov.w) > THRESHV) ? ov.w : 0.f;
      *(float4*)(out + off) = ov;
    }
  }
}

extern "C" void kernel_launch(void* const* d_in, const int* in_sizes, int n_in,
                              void* d_out, int out_size, void* d_ws, size_t ws_size,
                              hipStream_t stream) {
  const float* spike = (const float*)d_in[0];   // [B, T, D] fp32
  const float* pmem  = (const float*)d_in[1];   // [W, D]    fp32
  float*       outp  = (float*)d_out;           // [B, T, D] fp32
  (void)in_sizes; (void)n_in; (void)out_size; (void)d_ws; (void)ws_size;
  event_proc_kernel<<<TDIM, 256, 0, stream>>>(spike, pmem, outp);
}